// HungarianMatcher_11484742549831
// MI455X (gfx1250) — compile-verified
//
#include <hip/hip_runtime.h>
#include <stdint.h>

// Fused DETR-style Hungarian cost matrix for gfx1250 (MI455X).
//  - logits tile streamed global->LDS via GLOBAL_LOAD_ASYNC_TO_LDS_B128 (ASYNCcnt DMA)
//  - cost_class gather expressed as one-hot GEMM on V_WMMA_F32_16X16X4_F32 (exact f32)
//  - segment L1 + generalized 1D IoU fused in the epilogue on the WMMA accumulators

typedef float v2f __attribute__((ext_vector_type(2)));
typedef float v8f __attribute__((ext_vector_type(8)));

#define BS_   64
#define NQ_   300
#define NC_   200
#define NT_   32
#define QT_   16
#define ROWP_ 204   // padded LDS row stride in dwords (204 % 64 = 12 -> conflict-free over 16 rows; 816 B row = 16B-aligned)

__global__ __launch_bounds__(32)
void matcher_cost_kernel(const float* __restrict__ logits,     // [BS,NQ,NC]
                         const float* __restrict__ pred_seg,   // [BS,NQ,2]
                         const float* __restrict__ tseg_g,     // [BS,NT,2]
                         const float* __restrict__ lengths,    // [BS,1]
                         const int*   __restrict__ labels_g,   // [BS,NT]
                         float*       __restrict__ out)        // [BS,NQ,NT]
{
    __shared__ alignas(16) float lds_exp[QT_ * ROWP_];   // raw logits, then exp(logit - rowmax)
    __shared__ float lds_invsum[QT_];
    __shared__ float lds_oseg[QT_ * 2];
    __shared__ float lds_tseg[NT_ * 2];
    __shared__ int   lds_lab[NT_];

    const int b    = blockIdx.y;
    const int q0   = blockIdx.x * QT_;
    const int lane = threadIdx.x;     // 0..31 (wave32)
    const int qloc = lane & 15;       // query row within tile
    const int half = lane >> 4;       // 0/1: which class half this lane serves
    const int q    = q0 + qloc;
    const int cb   = half * 100;      // this lane covers classes [cb, cb+100)

    float* row = &lds_exp[qloc * ROWP_];

    // ---- async DMA: 16 rows x 200 classes, global -> LDS, no VGPR staging ----
    // Each lane copies 25 x 16B chunks of its (row, class-half). Partial last
    // tile: clamp source row in-bounds (results masked at store anyway).
    const int      qc    = (q < NQ_) ? q : (NQ_ - 1);
    const uint64_t gbase = (uint64_t)(uintptr_t)(logits + ((size_t)b * NQ_ + q0) * NC_);
    const uint32_t goff  = (uint32_t)(((qc - q0) * NC_ + cb) * sizeof(float));
    const uint32_t loff  = (uint32_t)(uintptr_t)(&lds_exp[0])           // LDS byte offset = addr[31:0]
                         + (uint32_t)((qloc * ROWP_ + cb) * sizeof(float));
#pragma unroll
    for (int i = 0; i < 25; ++i) {
        asm volatile("global_load_async_to_lds_b128 %0, %1, %2 offset:%3"
                     :: "v"(loff), "v"(goff), "s"(gbase), "i"(16 * i)
                     : "memory");
    }

    // ---- overlap the DMA with the small side-data loads ----
    if (half == 0) {
        if (q < NQ_) {
            lds_oseg[2 * qloc + 0] = pred_seg[((size_t)b * NQ_ + q) * 2 + 0];
            lds_oseg[2 * qloc + 1] = pred_seg[((size_t)b * NQ_ + q) * 2 + 1];
        } else {
            lds_oseg[2 * qloc + 0] = 0.0f;
            lds_oseg[2 * qloc + 1] = 0.0f;
        }
    }
    lds_tseg[2 * lane + 0] = tseg_g[((size_t)b * NT_ + lane) * 2 + 0];
    lds_tseg[2 * lane + 1] = tseg_g[((size_t)b * NT_ + lane) * 2 + 1];
    lds_lab[lane]          = labels_g[(size_t)b * NT_ + lane];

    asm volatile("s_wait_asynccnt 0x0" ::: "memory");   // logits tile resident in LDS

    // ---- softmax passes over own (row, half): max, then exp + sum ----
    float vmax = -3.402823466e38f;
#pragma unroll 4
    for (int i = 0; i < 100; ++i) vmax = fmaxf(vmax, row[cb + i]);
    vmax = fmaxf(vmax, __shfl_xor(vmax, 16, 32));       // combine the two halves

    float s = 0.0f;
#pragma unroll 4
    for (int i = 0; i < 100; ++i) {
        float e = __expf(row[cb + i] - vmax);
        row[cb + i] = e;                                 // unnormalized; scale after WMMA
        s += e;
    }
    s += __shfl_xor(s, 16, 32);
    if (half == 0) lds_invsum[qloc] = 1.0f / s;

    __syncthreads();

    // B-fragment labels: N = lane&15; tile0 -> targets 0..15, tile1 -> 16..31
    const int lab0 = lds_lab[qloc];
    const int lab1 = lds_lab[16 + qloc];

    // ---- one-hot GEMM: S[q,t] = sum_c exp[q,c] * (label_t == c) ----
    // A 16x4 f32 layout: lane l: M=l&15, K = vgpr + 2*half  (ISA 7.12.2)
    // B 4x16 f32 layout: lane l: N=l&15, K = vgpr + 2*half  (C/D-style row striping)
    v8f acc0 = {};
    v8f acc1 = {};
#pragma unroll 2
    for (int kb = 0; kb < NC_; kb += 4) {
        const int cc = kb + 2 * half;
        v2f a;
        a.x = lds_exp[qloc * ROWP_ + cc];
        a.y = lds_exp[qloc * ROWP_ + cc + 1];
        v2f b0, b1;
        b0.x = (lab0 == cc)     ? 1.0f : 0.0f;
        b0.y = (lab0 == cc + 1) ? 1.0f : 0.0f;
        b1.x = (lab1 == cc)     ? 1.0f : 0.0f;
        b1.y = (lab1 == cc + 1) ? 1.0f : 0.0f;
        // (neg_a, A, neg_b, B, c_mod, C, reuse_a, reuse_b)
        acc0 = __builtin_amdgcn_wmma_f32_16x16x4_f32(false, a, false, b0, (short)0, acc0, false, false);
        acc1 = __builtin_amdgcn_wmma_f32_16x16x4_f32(false, a, false, b1, (short)0, acc1, false, false);
    }

    // ---- epilogue: segment L1 + generalized 1D IoU, fused store ----
    const float L   = lengths[b];
    const float tc0 = lds_tseg[2 * qloc + 0],        tw0 = lds_tseg[2 * qloc + 1];
    const float tc1 = lds_tseg[2 * (16 + qloc) + 0], tw1 = lds_tseg[2 * (16 + qloc) + 1];
    const float ts0 = L * (tc0 - 0.5f * tw0), te0 = L * (tc0 + 0.5f * tw0);
    const float ts1 = L * (tc1 - 0.5f * tw1), te1 = L * (tc1 + 0.5f * tw1);

#pragma unroll
    for (int v = 0; v < 8; ++v) {
        const int m  = v + 8 * half;       // D layout: M = vgpr + 8*(lane>=16)
        const int qq = q0 + m;
        if (qq >= NQ_) continue;           // divergence only after all WMMAs
        const float oc   = lds_oseg[2 * m + 0];
        const float ow   = lds_oseg[2 * m + 1];
        const float invs = lds_invsum[m];
        const float os = L * (oc - 0.5f * ow), oe = L * (oc + 0.5f * ow);
        {   // targets 0..15 (t = qloc)
            float cseg  = fabsf(oc - tc0) + fabsf(ow - tw0);
            float inter = fmaxf(fminf(oe, te0) - fmaxf(os, ts0), 0.0f);
            float uni   = (oe - os) + (te0 - ts0) - inter;
            float enc   = fmaxf(oe, te0) - fminf(os, ts0);
            float giou  = inter / uni - (enc - uni) / enc;
            out[((size_t)b * NQ_ + qq) * NT_ + qloc] = cseg - acc0[v] * invs - giou;
        }
        {   // targets 16..31 (t = 16 + qloc)
            float cseg  = fabsf(oc - tc1) + fabsf(ow - tw1);
            float inter = fmaxf(fminf(oe, te1) - fmaxf(os, ts1), 0.0f);
            float uni   = (oe - os) + (te1 - ts1) - inter;
            float enc   = fmaxf(oe, te1) - fminf(os, ts1);
            float giou  = inter / uni - (enc - uni) / enc;
            out[((size_t)b * NQ_ + qq) * NT_ + 16 + qloc] = cseg - acc1[v] * invs - giou;
        }
    }
}

extern "C" void kernel_launch(void* const* d_in, const int* in_sizes, int n_in,
                              void* d_out, int out_size, void* d_ws, size_t ws_size,
                              hipStream_t stream) {
    const float* logits = (const float*)d_in[0];   // pred_logits [64,300,200]
    const float* pseg   = (const float*)d_in[1];   // pred_segments [64,300,2]
    const float* tseg   = (const float*)d_in[2];   // tgt_segments [64,32,2]
    const float* lens   = (const float*)d_in[3];   // lengths [64,1]
    const int*   labs   = (const int*)d_in[4];     // tgt_labels [64,32]
    float* out = (float*)d_out;                    // [64,300,32]

    (void)in_sizes; (void)n_in; (void)out_size; (void)d_ws; (void)ws_size;

    dim3 grid((NQ_ + QT_ - 1) / QT_, BS_);         // 19 x 64 blocks, 1 wave each
    matcher_cost_kernel<<<grid, 32, 0, stream>>>(logits, pseg, tseg, lens, labs, out);
}